// RetinaNetDecoder_31250182045896
// MI455X (gfx1250) — compile-verified
//
#include <hip/hip_runtime.h>
#include <hip/hip_bf16.h>

#define IMAGE_W 640
#define IMAGE_H 640
#define MIN_SCORE 0.05f
#define NMS_THR 0.5f
#define MAX_DET 100
#define NB 8
#define NA 76725
#define NC 80

typedef float v4f __attribute__((ext_vector_type(4)));

#define NEG_INF (-__builtin_inff())

// ---------------------------------------------------------------------------
// Kernel 1: per-anchor decode + class max/argmax + area. Bandwidth bound:
// ~233 MB moved => ~10us floor @ 23.3 TB/s. One thread per (batch, anchor).
// Class-score rows are streamed with non-temporal B128 loads (196 MB stream,
// don't pollute the 192 MB L2) plus a software prefetch of the next row.
// ---------------------------------------------------------------------------
__global__ __launch_bounds__(256) void retinanet_decode_kernel(
    const float* __restrict__ cls, const float* __restrict__ reg,
    const float* __restrict__ anc,
    float* __restrict__ work, float* __restrict__ clsout,
    float* __restrict__ area, float* __restrict__ boxes)
{
    const int tid = blockIdx.x * blockDim.x + threadIdx.x;
    if (tid >= NB * NA) return;

    const float* row = cls + (size_t)tid * NC;
    // streaming hint: pull the next row's line while we reduce this one
    __builtin_prefetch(row + NC, 0, 0);

    // max / argmax over 80 classes, first-occurrence tie-break (strict >)
    float best = -1.0f;
    int bestc = 0;
    const v4f* rowv = (const v4f*)row;
#pragma unroll
    for (int k = 0; k < NC / 4; ++k) {
        v4f v = __builtin_nontemporal_load(rowv + k);
        if (v.x > best) { best = v.x; bestc = 4 * k + 0; }
        if (v.y > best) { best = v.y; bestc = 4 * k + 1; }
        if (v.z > best) { best = v.z; bestc = 4 * k + 2; }
        if (v.w > best) { best = v.w; bestc = 4 * k + 3; }
    }

    v4f r  = *(const v4f*)(reg + (size_t)tid * 4);
    v4f a4 = *(const v4f*)(anc + (size_t)tid * 4);

    // snap_tx_ty_tw_th_to_x1_y1_x2_y2
    float w  = a4.z - a4.x;
    float h  = a4.w - a4.y;
    float cx = a4.x + 0.5f * w;
    float cy = a4.y + 0.5f * h;
    float pw = expf(r.z * 0.2f) * w;
    float ph = expf(r.w * 0.2f) * h;
    float pcx = r.x * 0.1f * w + cx;
    float pcy = r.y * 0.1f * h + cy;

    // truncate toward zero like astype(int32), then one-sided clamps
    int x1 = (int)(pcx - 0.5f * pw);
    int y1 = (int)(pcy - 0.5f * ph);
    int x2 = (int)(pcx + 0.5f * pw);
    int y2 = (int)(pcy + 0.5f * ph);
    x1 = max(x1, 0);
    y1 = max(y1, 0);
    x2 = min(x2, IMAGE_W - 1);
    y2 = min(y2, IMAGE_H - 1);

    v4f bx = { (float)x1, (float)y1, (float)x2, (float)y2 };
    *(v4f*)(boxes + (size_t)tid * 4) = bx;
    area[tid]   = (bx.z - bx.x) * (bx.w - bx.y);
    clsout[tid] = (float)bestc;
    work[tid]   = (best > MIN_SCORE) ? best : NEG_INF;
}

// ---------------------------------------------------------------------------
// Kernel 2: greedy NMS. One block (32 wave32 waves) per image; 100 strictly
// sequential picks. Block argmax = per-thread strided scan -> wave32 shuffle
// reduction -> LDS tree across 32 waves. Lowest-index tie-break reproduces
// jnp.argmax first-occurrence semantics.
// ---------------------------------------------------------------------------
__global__ __launch_bounds__(1024) void retinanet_nms_kernel(
    const float* __restrict__ cls_f, const float* __restrict__ area,
    const float* __restrict__ boxes, float* __restrict__ work,
    float* __restrict__ out_s, float* __restrict__ out_c,
    float* __restrict__ out_b)
{
    const int b = blockIdx.x;
    const float* A_ = area  + (size_t)b * NA;
    const float* C_ = cls_f + (size_t)b * NA;
    const float* BX = boxes + (size_t)b * NA * 4;
    float*       W  = work  + (size_t)b * NA;

    __shared__ float red_s[32];
    __shared__ int   red_i[32];
    __shared__ float bb[7];   // x1,y1,x2,y2,area,class,score
    __shared__ int   bidx;
    __shared__ int   done;

    const int t    = threadIdx.x;
    const int lane = t & 31;
    const int wave = t >> 5;

    if (t == 0) done = 0;
    __syncthreads();

    for (int i = 0; i < MAX_DET; ++i) {
        if (done) {
            if (t == 0) {
                out_s[b * MAX_DET + i] = -1.0f;
                out_c[b * MAX_DET + i] = -1.0f;
                v4f m1 = { -1.0f, -1.0f, -1.0f, -1.0f };
                *(v4f*)(out_b + ((size_t)b * MAX_DET + i) * 4) = m1;
            }
            continue;
        }

        // ---- block-wide argmax over work[] ----
        float bs = NEG_INF;
        int   bi = 0x7fffffff;
        for (int a = t; a < NA; a += 1024) {
            float s = W[a];
            if (s > bs || (s == bs && a < bi)) { bs = s; bi = a; }
        }
        // wave32 shuffle reduction
        for (int off = 16; off > 0; off >>= 1) {
            float os = __shfl_down(bs, off, 32);
            int   oi = __shfl_down(bi, off, 32);
            if (os > bs || (os == bs && oi < bi)) { bs = os; bi = oi; }
        }
        if (lane == 0) { red_s[wave] = bs; red_i[wave] = bi; }
        __syncthreads();
        if (wave == 0) {
            bs = red_s[lane];
            bi = red_i[lane];
            for (int off = 16; off > 0; off >>= 1) {
                float os = __shfl_down(bs, off, 32);
                int   oi = __shfl_down(bi, off, 32);
                if (os > bs || (os == bs && oi < bi)) { bs = os; bi = oi; }
            }
            if (lane == 0) {
                if (bs == NEG_INF) {
                    done = 1;
                } else {
                    bidx = bi;
                    v4f bx = *(const v4f*)(BX + (size_t)bi * 4);
                    bb[0] = bx.x; bb[1] = bx.y; bb[2] = bx.z; bb[3] = bx.w;
                    bb[4] = A_[bi];
                    bb[5] = C_[bi];
                    bb[6] = bs;
                    out_s[b * MAX_DET + i] = bs;
                    out_c[b * MAX_DET + i] = bb[5];
                    *(v4f*)(out_b + ((size_t)b * MAX_DET + i) * 4) = bx;
                }
            }
        }
        __syncthreads();

        if (done) {
            if (t == 0) {
                out_s[b * MAX_DET + i] = -1.0f;
                out_c[b * MAX_DET + i] = -1.0f;
                v4f m1 = { -1.0f, -1.0f, -1.0f, -1.0f };
                *(v4f*)(out_b + ((size_t)b * MAX_DET + i) * 4) = m1;
            }
            continue;
        }

        // ---- suppress same-class boxes with IoU >= thr ----
        const float x1 = bb[0], y1 = bb[1], x2 = bb[2], y2 = bb[3];
        const float ab = bb[4], cb = bb[5];
        const int   idx = bidx;
        for (int a = t; a < NA; a += 1024) {
            float s = W[a];
            if (a == idx) { W[a] = NEG_INF; continue; }   // unconditional kill of pick
            if (s == NEG_INF) continue;                   // already dead
            if (C_[a] != cb) continue;                    // per-class NMS: other classes kept
            v4f bx = *(const v4f*)(BX + (size_t)a * 4);
            float xx1 = fmaxf(bx.x, x1);
            float yy1 = fmaxf(bx.y, y1);
            float xx2 = fminf(bx.z, x2);
            float yy2 = fminf(bx.w, y2);
            float inter = fmaxf(xx2 - xx1, 0.0f) * fmaxf(yy2 - yy1, 0.0f);
            float uni   = A_[a] + ab - inter;
            float iou   = (uni > 0.0f) ? (inter / uni) : 0.0f;
            if (iou >= NMS_THR) W[a] = NEG_INF;
        }
        __syncthreads();
    }
}

// ---------------------------------------------------------------------------
// Launch: inputs are cls_heads [B,A,C], reg_heads [B,A,4], batch_anchors
// [B,A,4] (all fp32). Output is out_s [B,100] ++ out_c [B,100] ++
// out_b [B,100,4] flattened fp32. Workspace layout (floats):
//   [0,BA)    work scores (mutated by NMS)
//   [BA,2BA)  class (as float)
//   [2BA,3BA) area
//   [3BA,7BA) boxes x1,y1,x2,y2      (total 7*BA*4 B = 17.2 MB)
// ---------------------------------------------------------------------------
extern "C" void kernel_launch(void* const* d_in, const int* in_sizes, int n_in,
                              void* d_out, int out_size, void* d_ws, size_t ws_size,
                              hipStream_t stream) {
    const float* cls = (const float*)d_in[0];
    const float* reg = (const float*)d_in[1];
    const float* anc = (const float*)d_in[2];
    float* out = (float*)d_out;
    float* ws  = (float*)d_ws;

    const size_t BA = (size_t)NB * NA;
    float* work  = ws;
    float* clsf  = ws + BA;
    float* areaw = ws + 2 * BA;
    float* boxes = ws + 3 * BA;

    float* out_s = out;
    float* out_c = out + NB * MAX_DET;
    float* out_b = out + 2 * NB * MAX_DET;

    const int total = NB * NA;
    retinanet_decode_kernel<<<(total + 255) / 256, 256, 0, stream>>>(
        cls, reg, anc, work, clsf, areaw, boxes);
    retinanet_nms_kernel<<<NB, 1024, 0, stream>>>(
        clsf, areaw, boxes, work, out_s, out_c, out_b);
}